// Net_37005438222412
// MI455X (gfx1250) — compile-verified
//
#include <hip/hip_runtime.h>
#include <math.h>

// Problem constants (from reference)
#define N_NODES 100000
#define N_EDGES 1600000
#define FEAT    64
#define HID     64
#define NCLS    40

typedef __attribute__((ext_vector_type(2))) float v2f;
typedef __attribute__((ext_vector_type(8))) float v8f;

// ---------------------------------------------------------------------------
// Utility: zero fill
// ---------------------------------------------------------------------------
__global__ void fill_zero_kernel(float* __restrict__ p, long long n) {
    long long i = (long long)blockIdx.x * blockDim.x + threadIdx.x;
    if (i < n) p[i] = 0.0f;
}

// ---------------------------------------------------------------------------
// Degree accumulation: deg_out[row[e]] += 1 ; deg_in[col[e]] += 1
// ---------------------------------------------------------------------------
__global__ void degree_kernel(const int* __restrict__ row, const int* __restrict__ col,
                              float* __restrict__ deg_out, float* __restrict__ deg_in,
                              int E) {
    int e = blockIdx.x * blockDim.x + threadIdx.x;
    if (e >= E) return;
    __hip_atomic_fetch_add(&deg_out[row[e]], 1.0f, __ATOMIC_RELAXED, __HIP_MEMORY_SCOPE_AGENT);
    __hip_atomic_fetch_add(&deg_in [col[e]], 1.0f, __ATOMIC_RELAXED, __HIP_MEMORY_SCOPE_AGENT);
}

// In place: v -> (v > 0) ? rsqrt(v) : 0   (deg >= 1 when > 0, so max(v,1)==v)
__global__ void inv_sqrt_kernel(float* __restrict__ p, int n) {
    int i = blockIdx.x * blockDim.x + threadIdx.x;
    if (i < n) {
        float v = p[i];
        p[i] = (v > 0.0f) ? rsqrtf(v) : 0.0f;
    }
}

// ---------------------------------------------------------------------------
// Edge weights (shared by all three layers): w[e] = inv_out[row]*inv_in[col]
// ---------------------------------------------------------------------------
__global__ void edge_weight_kernel(const int* __restrict__ row, const int* __restrict__ col,
                                   const float* __restrict__ inv_out,
                                   const float* __restrict__ inv_in,
                                   float* __restrict__ w, int E) {
    int e = blockIdx.x * blockDim.x + threadIdx.x;
    if (e >= E) return;
    w[e] = inv_out[row[e]] * inv_in[col[e]];
}

// ---------------------------------------------------------------------------
// SpMM scatter:  Y[row[e], :] += w[e] * act(X[col[e], :])
// One thread per (edge, feature). 64 features -> 2 waves per edge; the X
// gather and the atomic scatter are both 256B-contiguous per edge.
// ---------------------------------------------------------------------------
__global__ void spmm_scatter_kernel(const float* __restrict__ X,
                                    const int* __restrict__ row,
                                    const int* __restrict__ col,
                                    const float* __restrict__ wgt,
                                    float* __restrict__ Y,
                                    int E, int relu) {
    long long t = (long long)blockIdx.x * blockDim.x + threadIdx.x;
    int e = (int)(t >> 6);
    int f = (int)(t & 63);
    if (e >= E) return;
    int r = row[e];
    int c = col[e];
    float w = wgt[e];
    float v = X[(size_t)c * 64 + f];
    if (relu) v = fmaxf(v, 0.0f);
    __hip_atomic_fetch_add(&Y[(size_t)r * 64 + f], w * v,
                           __ATOMIC_RELAXED, __HIP_MEMORY_SCOPE_AGENT);
}

// ---------------------------------------------------------------------------
// Dense GEMM with fp32 WMMA:  Z[nrows x ncols] = Y[nrows x 64] @ W[64 x ncols] + bias
// NT = number of 16-wide N tiles (4 for 64 cols, 3 for 40 cols padded to 48).
//
// V_WMMA_F32_16X16X4_F32 operand layouts (wave32):
//   A 16x4 : lane l holds row M=l%16; VGPR r holds K = (l/16)*2 + r
//   B 4x16 : lane l holds col N=l%16; VGPR r holds K = (l/16)*2 + r
//   C 16x16: lane l col N=l%16; VGPR i row M = (l/16)*8 + i
//
// W is staged into LDS pre-swizzled into per-(kt,nt) fragments so each lane's
// B operand is a single 8B ds_load_b64 into an even-aligned VGPR pair
// (conflict-free: 32 lanes x 8B hit all 64 banks exactly once).
// ---------------------------------------------------------------------------
template <int NT>
__global__ void gemm_wmma_kernel(const float* __restrict__ Y,
                                 const float* __restrict__ W,
                                 const float* __restrict__ bias,
                                 float* __restrict__ Z,
                                 int nrows, int ncols, int ldz) {
    // 16 K-tiles * NT N-tiles, each fragment = 32 lanes * 2 floats
    __shared__ float sW[16 * NT * 64];

    const int tid = threadIdx.x;
    for (int idx = tid; idx < 16 * NT * 64; idx += blockDim.x) {
        int frag = idx >> 6;        // kt*NT + nt
        int lr   = idx & 63;        // lane*2 + r
        int ln   = lr >> 1;
        int r    = lr & 1;
        int kt   = frag / NT;
        int nt   = frag % NT;
        int kk   = kt * 4 + (ln >> 4) * 2 + r;   // K index 0..63
        int cc   = nt * 16 + (ln & 15);          // N index
        sW[idx] = (cc < ncols) ? W[kk * ncols + cc] : 0.0f;
    }
    __syncthreads();

    const int wave = tid >> 5;
    const int lane = tid & 31;
    const int tile = blockIdx.x * (blockDim.x >> 5) + wave;
    if (tile >= nrows / 16) return;

    const int r0 = tile * 16;
    const int m  = lane & 15;   // M for A-frag / N for B,C-frags
    const int kh = lane >> 4;   // lane-half selects K offset {0,1} vs {2,3}

    v8f c[NT];
#pragma unroll
    for (int nt = 0; nt < NT; ++nt) c[nt] = (v8f){0, 0, 0, 0, 0, 0, 0, 0};

    const float* arow  = Y + (size_t)(r0 + m) * 64 + kh * 2;
    const float* bfrag = sW + lane * 2;

#pragma unroll
    for (int kt = 0; kt < 16; ++kt) {            // K = 64 in steps of 4
        v2f a = *(const v2f*)(arow + kt * 4);    // global_load_b64
#pragma unroll
        for (int nt = 0; nt < NT; ++nt) {
            v2f b = *(const v2f*)(bfrag + (kt * NT + nt) * 64);  // ds_load_b64
            c[nt] = __builtin_amdgcn_wmma_f32_16x16x4_f32(
                false, a, false, b, (short)0, c[nt], false, false);
        }
    }

#pragma unroll
    for (int nt = 0; nt < NT; ++nt) {
        int ncol = nt * 16 + m;
        if (ncol >= ncols) continue;
        float bb = bias[ncol];
#pragma unroll
        for (int i = 0; i < 8; ++i) {
            int rr = r0 + kh * 8 + i;
            Z[(size_t)rr * ldz + ncol] = c[nt][i] + bb;
        }
    }
}

// ---------------------------------------------------------------------------
// Row-wise log-softmax over NCLS=40 columns; one wave32 per row.
// ---------------------------------------------------------------------------
__global__ void log_softmax_kernel(const float* __restrict__ O,
                                   float* __restrict__ OUT, int nrows) {
    int wave  = threadIdx.x >> 5;
    int lane  = threadIdx.x & 31;
    int rowid = blockIdx.x * (blockDim.x >> 5) + wave;
    if (rowid >= nrows) return;

    const float* rp = O + (size_t)rowid * NCLS;
    float v0 = (lane < NCLS)        ? rp[lane]      : -INFINITY;
    float v1 = (lane + 32 < NCLS)   ? rp[lane + 32] : -INFINITY;

    float mx = fmaxf(v0, v1);
    for (int off = 16; off >= 1; off >>= 1)
        mx = fmaxf(mx, __shfl_xor(mx, off, 32));

    float s = 0.0f;
    if (lane < NCLS)      s += expf(v0 - mx);
    if (lane + 32 < NCLS) s += expf(v1 - mx);
    for (int off = 16; off >= 1; off >>= 1)
        s += __shfl_xor(s, off, 32);

    float lg = logf(s);
    float* op = OUT + (size_t)rowid * NCLS;
    if (lane < NCLS)      op[lane]      = v0 - mx - lg;
    if (lane + 32 < NCLS) op[lane + 32] = v1 - mx - lg;
}

// ---------------------------------------------------------------------------
// Orchestration.
// Inputs (setup_inputs order):
//   0:x  1:edge_index  2:W1s 3:b1s 4:W1d 5:b1d 6:W2s 7:b2s 8:W2d 9:b2d
//   10:Wos 11:bos 12:Wod 13:bod        (ALPHA=1 -> *d weights unused)
// Output: d_out = [ h (N*64 floats) | log_softmax(out) (N*40 floats) ]
// ---------------------------------------------------------------------------
extern "C" void kernel_launch(void* const* d_in, const int* in_sizes, int n_in,
                              void* d_out, int out_size, void* d_ws, size_t ws_size,
                              hipStream_t stream) {
    const float* x   = (const float*)d_in[0];
    const int*   ei  = (const int*)d_in[1];
    const int*   row = ei;             // edge_index[0, :]
    const int*   col = ei + N_EDGES;   // edge_index[1, :]
    const float* W1s = (const float*)d_in[2];
    const float* b1s = (const float*)d_in[3];
    const float* W2s = (const float*)d_in[6];
    const float* b2s = (const float*)d_in[7];
    const float* Wos = (const float*)d_in[10];
    const float* bos = (const float*)d_in[11];

    float* out_h  = (float*)d_out;                       // N*64
    float* out_ls = out_h + (size_t)N_NODES * HID;       // N*40

    // Workspace layout (floats):
    //   inv_out[N] inv_in[N] bufY[N*64] bufZ[N*64] wgt[E]
    float* inv_out = (float*)d_ws;
    float* inv_in  = inv_out + N_NODES;
    float* bufY    = inv_in + N_NODES;
    float* bufZ    = bufY + (size_t)N_NODES * 64;
    float* wgt     = bufZ + (size_t)N_NODES * 64;

    const long long yn = (long long)N_NODES * 64;
    const int fill_deg_blocks = (2 * N_NODES + 255) / 256;
    const int fill_y_blocks   = (int)((yn + 255) / 256);
    const int edge_blocks     = (N_EDGES + 255) / 256;
    const long long spmm_thr  = (long long)N_EDGES * 64;
    const int spmm_blocks     = (int)((spmm_thr + 255) / 256);
    const int row_tiles       = N_NODES / 16;            // 6250 (exact)
    const int gemm_blocks     = (row_tiles + 7) / 8;     // 8 waves per block
    const int ls_blocks       = (N_NODES + 7) / 8;       // 8 rows per block

    // --- normalization weights (once, shared by all layers) ---
    fill_zero_kernel<<<fill_deg_blocks, 256, 0, stream>>>(inv_out, 2 * N_NODES);
    degree_kernel<<<edge_blocks, 256, 0, stream>>>(row, col, inv_out, inv_in, N_EDGES);
    inv_sqrt_kernel<<<fill_deg_blocks, 256, 0, stream>>>(inv_out, 2 * N_NODES);
    edge_weight_kernel<<<edge_blocks, 256, 0, stream>>>(row, col, inv_out, inv_in,
                                                        wgt, N_EDGES);

    // --- layer 1: x0 = (A @ x) @ W1s + b1s ---
    fill_zero_kernel<<<fill_y_blocks, 256, 0, stream>>>(bufY, yn);
    spmm_scatter_kernel<<<spmm_blocks, 256, 0, stream>>>(x, row, col, wgt,
                                                         bufY, N_EDGES, 0);
    gemm_wmma_kernel<4><<<gemm_blocks, 256, 0, stream>>>(bufY, W1s, b1s, bufZ,
                                                         N_NODES, HID, HID);

    // --- layer 2: h = (A @ relu(x0)) @ W2s + b2s   (h is first output) ---
    fill_zero_kernel<<<fill_y_blocks, 256, 0, stream>>>(bufY, yn);
    spmm_scatter_kernel<<<spmm_blocks, 256, 0, stream>>>(bufZ, row, col, wgt,
                                                         bufY, N_EDGES, 1);
    gemm_wmma_kernel<4><<<gemm_blocks, 256, 0, stream>>>(bufY, W2s, b2s, out_h,
                                                         N_NODES, HID, HID);

    // --- layer 3: out = (A @ relu(h)) @ Wos + bos ; then log_softmax ---
    fill_zero_kernel<<<fill_y_blocks, 256, 0, stream>>>(bufY, yn);
    spmm_scatter_kernel<<<spmm_blocks, 256, 0, stream>>>(out_h, row, col, wgt,
                                                         bufY, N_EDGES, 1);
    gemm_wmma_kernel<3><<<gemm_blocks, 256, 0, stream>>>(bufY, Wos, bos, bufZ,
                                                         N_NODES, NCLS, NCLS);
    log_softmax_kernel<<<ls_blocks, 256, 0, stream>>>(bufZ, out_ls, N_NODES);
}